// GatP3First_17437567221934
// MI455X (gfx1250) — compile-verified
//
#include <hip/hip_runtime.h>
#include <hip/hip_bf16.h>

#define N_FEAT 128
#define HEADS 4
#define F_HEAD 32
#define NEG_SLOPE 0.2f

typedef __attribute__((ext_vector_type(2))) float v2f;
typedef __attribute__((ext_vector_type(8))) float v8f;

// ---------------------------------------------------------------------------
// 1) init: out = bias broadcast, m_enc = enc(-inf), s = 0
// ---------------------------------------------------------------------------
__global__ __launch_bounds__(256) void gat_init(float* __restrict__ out,
                                                unsigned* __restrict__ m_enc,
                                                float* __restrict__ s,
                                                const float* __restrict__ bias,
                                                int n) {
  long long idx = (long long)blockIdx.x * blockDim.x + threadIdx.x;
  long long total = (long long)n * N_FEAT;
  if (idx < total) out[idx] = bias[(int)(idx & (N_FEAT - 1))];
  if (idx < (long long)n * HEADS) {
    m_enc[idx] = 0x007FFFFFu;  // enc(-inf) = ~(0xFF800000)
    s[idx] = 0.0f;
  }
}

// ---------------------------------------------------------------------------
// 2) ft = feat @ W via V_WMMA_F32_16X16X4_F32.
//    Block = 256 threads = 8 waves; block covers 16 rows, wave w covers cols
//    [16w, 16w+16). 32 k-steps of K=4 each.
//    A 16x4 f32 layout:  lanes 0-15: v0=K0,v1=K1 ; lanes 16-31: v0=K2,v1=K3
//    B 4x16 f32 layout:  lanes 0-15: v0=K0,v1=K1 rows, N=lane ; lanes 16-31: K2,K3
//    C/D 16x16 f32:      VGPR v -> row v (lanes 0-15) / row v+8 (lanes 16-31)
// ---------------------------------------------------------------------------
__global__ __launch_bounds__(256) void gat_gemm_wmma(const float* __restrict__ feat,
                                                     const float* __restrict__ W,
                                                     float* __restrict__ ft) {
  const int lane   = threadIdx.x & 31;
  const int wave   = threadIdx.x >> 5;
  const int lane16 = lane & 15;
  const int kOff   = (lane < 16) ? 0 : 2;
  const int rowBase = blockIdx.x * 16;
  const int colBase = wave * 16;

  const float* aRow = feat + (size_t)(rowBase + lane16) * N_FEAT;
  const float* bCol = W + colBase + lane16;

  v8f acc = {};
#pragma unroll
  for (int k0 = 0; k0 < N_FEAT; k0 += 4) {
    v2f a, b;
    a.x = aRow[k0 + kOff + 0];
    a.y = aRow[k0 + kOff + 1];
    b.x = bCol[(size_t)(k0 + kOff + 0) * N_FEAT];
    b.y = bCol[(size_t)(k0 + kOff + 1) * N_FEAT];
    acc = __builtin_amdgcn_wmma_f32_16x16x4_f32(
        /*neg_a=*/false, a, /*neg_b=*/false, b,
        /*c_mod=*/(short)0, acc, /*reuse_a=*/false, /*reuse_b=*/false);
  }

  const int rowOff = (lane < 16) ? 0 : 8;
#pragma unroll
  for (int v = 0; v < 8; ++v) {
    ft[(size_t)(rowBase + v + rowOff) * N_FEAT + colBase + lane16] = acc[v];
  }
}

// ---------------------------------------------------------------------------
// 3) per-(node, head) attention logits: el = ft . attn_l, er = ft . attn_r
// ---------------------------------------------------------------------------
__global__ __launch_bounds__(256) void gat_logits(const float* __restrict__ ft,
                                                  const float* __restrict__ attn_l,
                                                  const float* __restrict__ attn_r,
                                                  float* __restrict__ el,
                                                  float* __restrict__ er, int n) {
  int idx = blockIdx.x * blockDim.x + threadIdx.x;
  if (idx >= n * HEADS) return;
  const int node = idx >> 2;
  const int h    = idx & 3;
  const float4* f  = (const float4*)(ft + (size_t)node * N_FEAT + h * F_HEAD);
  const float4* al = (const float4*)(attn_l + h * F_HEAD);
  const float4* ar = (const float4*)(attn_r + h * F_HEAD);
  float sl = 0.f, sr = 0.f;
#pragma unroll
  for (int i = 0; i < F_HEAD / 4; ++i) {
    float4 v = f[i], l = al[i], r = ar[i];
    sl += v.x * l.x + v.y * l.y + v.z * l.z + v.w * l.w;
    sr += v.x * r.x + v.y * r.y + v.z * r.z + v.w * r.w;
  }
  el[idx] = sl;
  er[idx] = sr;
}

// ---------------------------------------------------------------------------
// 4) per-edge leaky-relu score + segment max via order-preserving uint atomicMax
// ---------------------------------------------------------------------------
__device__ __forceinline__ unsigned enc_f32(float x) {
  unsigned u = __float_as_uint(x);
  return (u & 0x80000000u) ? ~u : (u | 0x80000000u);
}

__global__ __launch_bounds__(256) void gat_edge_scores(const float* __restrict__ el,
                                                       const float* __restrict__ er,
                                                       const int* __restrict__ src,
                                                       const int* __restrict__ dst,
                                                       float* __restrict__ e,
                                                       unsigned* __restrict__ m_enc,
                                                       int nEdges) {
  int idx = blockIdx.x * blockDim.x + threadIdx.x;
  if (idx >= nEdges * HEADS) return;
  const int ed = idx >> 2;
  const int h  = idx & 3;
  const int si = src[ed];
  const int di = dst[ed];
  float v = el[si * HEADS + h] + er[di * HEADS + h];
  v = (v > 0.f) ? v : v * NEG_SLOPE;
  e[idx] = v;
  atomicMax(&m_enc[di * HEADS + h], enc_f32(v));
}

// ---------------------------------------------------------------------------
// 5) decode m, guard non-finite -> 0 (reuse buffer as float)
// ---------------------------------------------------------------------------
__global__ __launch_bounds__(256) void gat_final_m(unsigned* __restrict__ m_enc, int n) {
  int idx = blockIdx.x * blockDim.x + threadIdx.x;
  if (idx >= n * HEADS) return;
  unsigned ev = m_enc[idx];
  unsigned u = (ev & 0x80000000u) ? (ev ^ 0x80000000u) : ~ev;
  if ((u & 0x7F800000u) == 0x7F800000u) u = 0u;  // inf/nan -> 0.0f
  m_enc[idx] = u;                                 // now holds float bits
}

// ---------------------------------------------------------------------------
// 6) a = exp(e - m[dst]); segment sum s[dst] += a   (a overwrites e in place)
// ---------------------------------------------------------------------------
__global__ __launch_bounds__(256) void gat_edge_exp(float* __restrict__ e,
                                                    const float* __restrict__ m,
                                                    const int* __restrict__ dst,
                                                    float* __restrict__ s, int nEdges) {
  int idx = blockIdx.x * blockDim.x + threadIdx.x;
  if (idx >= nEdges * HEADS) return;
  const int ed = idx >> 2;
  const int h  = idx & 3;
  const int di = dst[ed];
  float a = __expf(e[idx] - m[di * HEADS + h]);
  e[idx] = a;
  atomicAdd(&s[di * HEADS + h], a);
}

// ---------------------------------------------------------------------------
// 7) weighted scatter-sum: out[dst] += ft[src] * alpha. One thread per
//    (edge, feature); 128 contiguous lanes -> coalesced gather + f32 atomics.
// ---------------------------------------------------------------------------
__global__ __launch_bounds__(256) void gat_scatter(const float* __restrict__ ft,
                                                   const float* __restrict__ a,
                                                   const float* __restrict__ s,
                                                   const int* __restrict__ src,
                                                   const int* __restrict__ dst,
                                                   float* __restrict__ out,
                                                   int nEdges) {
  long long idx = (long long)blockIdx.x * blockDim.x + threadIdx.x;
  if (idx >= (long long)nEdges * N_FEAT) return;
  const int ed = (int)(idx >> 7);
  const int c  = (int)(idx & (N_FEAT - 1));
  const int h  = c >> 5;
  const int si = src[ed];
  const int di = dst[ed];
  float sv = s[di * HEADS + h];
  float alpha = a[ed * HEADS + h] / ((sv > 0.f) ? sv : 1.f);
  atomicAdd(&out[(size_t)di * N_FEAT + c], ft[(size_t)si * N_FEAT + c] * alpha);
}

// ---------------------------------------------------------------------------
extern "C" void kernel_launch(void* const* d_in, const int* in_sizes, int n_in,
                              void* d_out, int out_size, void* d_ws, size_t ws_size,
                              hipStream_t stream) {
  const float* feat   = (const float*)d_in[0];
  const float* W      = (const float*)d_in[1];
  const float* attn_l = (const float*)d_in[2];
  const float* attn_r = (const float*)d_in[3];
  const float* bias   = (const float*)d_in[4];
  const int*   src    = (const int*)d_in[5];
  const int*   dst    = (const int*)d_in[6];

  const int N = in_sizes[0] / N_FEAT;   // 50000
  const int E = in_sizes[5];            // 1600000

  // workspace layout
  char* ws = (char*)d_ws;
  float* ft = (float*)ws;                 ws += (size_t)N * N_FEAT * sizeof(float);
  float* el = (float*)ws;                 ws += (size_t)N * HEADS * sizeof(float);
  float* er = (float*)ws;                 ws += (size_t)N * HEADS * sizeof(float);
  unsigned* m_enc = (unsigned*)ws;        ws += (size_t)N * HEADS * sizeof(unsigned);
  float* s = (float*)ws;                  ws += (size_t)N * HEADS * sizeof(float);
  float* e = (float*)ws;                  ws += (size_t)E * HEADS * sizeof(float);

  float* out = (float*)d_out;

  const int TB = 256;

  // 1) init (out=bias, m=-inf encoded, s=0)
  {
    long long total = (long long)N * N_FEAT;
    int blocks = (int)((total + TB - 1) / TB);
    gat_init<<<blocks, TB, 0, stream>>>(out, m_enc, s, bias, N);
  }
  // 2) WMMA GEMM: ft = feat @ W   (N divisible by 16: 50000/16 = 3125)
  {
    int blocks = N / 16;
    gat_gemm_wmma<<<blocks, TB, 0, stream>>>(feat, W, ft);
  }
  // 3) logits
  {
    int blocks = (N * HEADS + TB - 1) / TB;
    gat_logits<<<blocks, TB, 0, stream>>>(ft, attn_l, attn_r, el, er, N);
  }
  // 4) edge scores + segment max
  {
    int blocks = (E * HEADS + TB - 1) / TB;
    gat_edge_scores<<<blocks, TB, 0, stream>>>(el, er, src, dst, e, m_enc, E);
  }
  // 5) finalize m
  {
    int blocks = (N * HEADS + TB - 1) / TB;
    gat_final_m<<<blocks, TB, 0, stream>>>(m_enc, N);
  }
  // 6) exp + segment sum
  {
    int blocks = (E * HEADS + TB - 1) / TB;
    gat_edge_exp<<<blocks, TB, 0, stream>>>(e, (const float*)m_enc, dst, s, E);
  }
  // 7) weighted scatter-sum
  {
    long long total = (long long)E * N_FEAT;
    int blocks = (int)((total + TB - 1) / TB);
    gat_scatter<<<blocks, TB, 0, stream>>>(ft, e, s, src, dst, out, E);
  }
}